// Interpolator_80771154969429
// MI455X (gfx1250) — compile-verified
//
#include <hip/hip_runtime.h>

typedef __attribute__((ext_vector_type(2))) float v2f;
typedef __attribute__((ext_vector_type(8))) float v8f;

#define TILES_PER_WAVE 8   // 8 tiles x 256 outputs = 2048 outputs per wave

// Polyphase FIR interpolator (factor 8, 128 taps) as fp32 WMMA GEMM tiles.
//   y[16u+q] = sum_j g_q[j] * x[2u - 7 + j]
//   g_q[j]   = 8 * h[8*(j - (q>=8)) + 7 - (q&7)],  zero outside j in [q>=8, 16+(q>=8))
// Per 16-group tile: D[16x16] = A[16x20] * B[20x16] via 5x v_wmma_f32_16x16x4_f32.
__global__ __launch_bounds__(256) void Interpolator_80771154969429_kernel(
    const float* __restrict__ xr,   // [Brows, N] real part
    const float* __restrict__ xi,   // [Brows, N] imag part
    const float* __restrict__ h,    // [128] FIR taps
    float* __restrict__ out,        // [2, Brows, N*8]
    int Brows, int N, int tilesPerRow)
{
    const int lane        = threadIdx.x & 31;
    const int w           = blockIdx.x * (blockDim.x >> 5) + (threadIdx.x >> 5);
    const int wavesPerRow = tilesPerRow / TILES_PER_WAVE;
    const int row         = w / wavesPerRow;          // wave-uniform
    if (row >= 2 * Brows) return;
    const int widx  = w - row * wavesPerRow;
    const int tile0 = widx * TILES_PER_WAVE;

    const long long M = (long long)N * 8;
    const float* __restrict__ x =
        (row < Brows ? xr + (long long)row * N
                     : xi + (long long)(row - Brows) * N);
    float* __restrict__ yrow = out + (long long)row * M;

    const int q  = lane & 15;       // B/D column (and A row index)
    const int hi = lane >> 4;       // upper lane half -> K/row offset
    const int kk = hi * 2;

    // ---- B matrix (wave-invariant), branchless clamp+select gather of h ----
    // lane L, chunk c, VGPR v: B[4c + kk + v][q] = g_q[4c + kk + v]
    const int p    = q & 7;
    const int joff = (q < 8) ? 0 : 1;
    v2f bm[5];
#pragma unroll
    for (int c = 0; c < 5; ++c) {
        const int jj0 = 4 * c + kk - joff;
        const int jj1 = jj0 + 1;
        const int h0  = min(max(8 * jj0 + 7 - p, 0), 127);   // always-safe address
        const int h1  = min(max(8 * jj1 + 7 - p, 0), 127);
        const float v0 = h[h0];
        const float v1 = h[h1];
        bm[c].x = ((unsigned)jj0 < 16u) ? 8.0f * v0 : 0.0f;
        bm[c].y = ((unsigned)jj1 < 16u) ? 8.0f * v1 : 0.0f;
    }

    // Per-lane A base index: A[r=q][kk+v] for tile t, chunk c lives at
    //   x[abase0 + 32*t + 4*c + v]
    const int abase0 = 32 * tile0 + 2 * q - 7 + kk;
    // Per-lane output base: tile t, D-VGPR v -> yrow[ybase0 + 256*t + 16*v]
    float* __restrict__ py = yrow + (long long)tile0 * 256 + 128 * hi + q;

    // Interior waves never touch indices <0 or >=N (tile 0 / last tile only).
    const int interior = (widx != 0) & (widx != wavesPerRow - 1);

    if (__builtin_amdgcn_readfirstlane(interior)) {
        // ---------- fast path: unconditional loads, immediate offsets ----------
        const float* __restrict__ px = x + abase0;
#pragma unroll
        for (int t = 0; t < TILES_PER_WAVE; ++t) {
            v8f acc = {0.f, 0.f, 0.f, 0.f, 0.f, 0.f, 0.f, 0.f};
#pragma unroll
            for (int c = 0; c < 5; ++c) {
                v2f a;
                a.x = px[32 * t + 4 * c];
                a.y = px[32 * t + 4 * c + 1];
                acc = __builtin_amdgcn_wmma_f32_16x16x4_f32(
                    false, a, false, bm[c], (short)0, acc, false, false);
            }
#pragma unroll
            for (int v = 0; v < 8; ++v)
                __builtin_nontemporal_store(acc[v], &py[256 * t + 16 * v]);
        }
    } else {
        // ---------- edge path: clamped address + select (no exec branches) ----------
#pragma unroll
        for (int t = 0; t < TILES_PER_WAVE; ++t) {
            v8f acc = {0.f, 0.f, 0.f, 0.f, 0.f, 0.f, 0.f, 0.f};
#pragma unroll
            for (int c = 0; c < 5; ++c) {
                const int i0 = abase0 + 32 * t + 4 * c;
                const int i1 = i0 + 1;
                const int c0 = min(max(i0, 0), N - 1);
                const int c1 = min(max(i1, 0), N - 1);
                const float v0 = x[c0];
                const float v1 = x[c1];
                v2f a;
                a.x = ((unsigned)i0 < (unsigned)N) ? v0 : 0.0f;  // SAME zero-pad
                a.y = ((unsigned)i1 < (unsigned)N) ? v1 : 0.0f;
                acc = __builtin_amdgcn_wmma_f32_16x16x4_f32(
                    false, a, false, bm[c], (short)0, acc, false, false);
            }
#pragma unroll
            for (int v = 0; v < 8; ++v)
                __builtin_nontemporal_store(acc[v], &py[256 * t + 16 * v]);
        }
    }
}

extern "C" void kernel_launch(void* const* d_in, const int* in_sizes, int n_in,
                              void* d_out, int out_size, void* d_ws, size_t ws_size,
                              hipStream_t stream) {
    const float* xr = (const float*)d_in[0];   // x_real [64, 32768]
    const float* xi = (const float*)d_in[1];   // x_imag [64, 32768]
    const float* h  = (const float*)d_in[2];   // fir_filter [128]
    // d_in[3] = factor (device scalar); kernel assumes factor=8, 128 taps.

    const int Brows = 64;
    const int N     = in_sizes[0] / Brows;            // 32768
    const int tilesPerRow = (N * 8) / 256;            // 1024
    const int wavesPerRow = tilesPerRow / TILES_PER_WAVE;   // 128
    const int totalWaves  = 2 * Brows * wavesPerRow;  // 16384
    const int wavesPerBlk = 8;                        // 256 threads, wave32
    const int blocks      = (totalWaves + wavesPerBlk - 1) / wavesPerBlk;

    Interpolator_80771154969429_kernel<<<blocks, 256, 0, stream>>>(
        xr, xi, h, (float*)d_out, Brows, N, tilesPerRow);
}